// GraphAttentionLayer_35304631173349
// MI455X (gfx1250) — compile-verified
//
#include <hip/hip_runtime.h>
#include <hip/hip_bf16.h>

// GAT layer for MI455X (gfx1250, wave32).
// N=100000 nodes, E=1600000 edges, C=64, H=4, D=16.

typedef __attribute__((ext_vector_type(2))) float v2f;
typedef __attribute__((ext_vector_type(8))) float v8f;

#define GAT_N 100000
#define GAT_E 1600000
#define GAT_C 64
#define GAT_H 4
#define GAT_D 16

// ---------------------------------------------------------------------------
// Fused QKV projection: q|k|v[n,:] = x[n,:] @ {Wq,Wk,Wv}, all [64x64] weights.
// One block = 128 threads = 4 wave32s; block handles a 16-row strip of x.
// Wave w computes the 16x16 output tile at columns [16w,16w+16) for q, k, v
// using V_WMMA_F32_16X16X4_F32 (exact fp32, K=4 per op, 16 ops per tile).
// ---------------------------------------------------------------------------
#define XS_STRIDE 68  // 64 + 4 pad: kills 16-way LDS bank conflict on col reads

__global__ __launch_bounds__(128) void gat_qkv_wmma(
    const float* __restrict__ x,
    const float* __restrict__ Wq,
    const float* __restrict__ Wk,
    const float* __restrict__ Wv,
    float* __restrict__ q,
    float* __restrict__ k,
    float* __restrict__ v)
{
    __shared__ float xs[16 * XS_STRIDE];

    const int row0 = blockIdx.x * 16;
    const int t    = threadIdx.x;

    // Cooperative load of the 16x64 fp32 x tile (8 floats per thread, b128).
    {
        const int r = t >> 3;          // 0..15
        const int c = (t & 7) * 8;     // 0,8,..,56
        const float4* src = reinterpret_cast<const float4*>(x + (row0 + r) * GAT_C + c);
        float4 a0 = src[0];
        float4 a1 = src[1];
        *reinterpret_cast<float4*>(&xs[r * XS_STRIDE + c])     = a0;
        *reinterpret_cast<float4*>(&xs[r * XS_STRIDE + c + 4]) = a1;
    }
    __syncthreads();

    const int wave = t >> 5;          // 0..3 -> output column tile
    const int lane = t & 31;
    const int n0   = wave * 16;

    // A-matrix fragments (16x4 fp32 per step): lane L<16 holds row M=L, K=k0,k0+1;
    // lanes 16..31 hold K=k0+2,k0+3.  Loaded once, reused for all three GEMMs.
    const int arow = lane & 15;
    const int koff = (lane >> 4) * 2;
    v2f afrag[16];
#pragma unroll
    for (int kk = 0; kk < 16; ++kk) {
        afrag[kk].x = xs[arow * XS_STRIDE + kk * 4 + koff];
        afrag[kk].y = xs[arow * XS_STRIDE + kk * 4 + koff + 1];
    }

    const int bn = n0 + (lane & 15);
    const int bk = (lane >> 4) * 2;
    const int om = (lane >> 4) * 8;   // C/D layout: VGPR r -> M = om + r
    const int on = lane & 15;

    const float* Ws[3]  = {Wq, Wk, Wv};
    float*       Os[3]  = {q, k, v};

#pragma unroll
    for (int m = 0; m < 3; ++m) {
        const float* __restrict__ W = Ws[m];
        v8f acc = {};
#pragma unroll
        for (int kk = 0; kk < 16; ++kk) {
            // B-matrix 4x16: row K striped across lanes within a VGPR.
            v2f b;
            b.x = W[(kk * 4 + bk) * GAT_C + bn];
            b.y = W[(kk * 4 + bk + 1) * GAT_C + bn];
            acc = __builtin_amdgcn_wmma_f32_16x16x4_f32(
                false, afrag[kk], false, b, (short)0, acc, false, false);
        }
        float* __restrict__ O = Os[m];
#pragma unroll
        for (int r = 0; r < 8; ++r)
            O[(row0 + om + r) * GAT_C + n0 + on] = acc[r];
    }
}

// ---------------------------------------------------------------------------
// Generic 16x16-tiled WMMA GEMM: out[n,:] = in[n,:] @ W  (W is 64x64).
// Same structure as above, single weight matrix. Used for the Wo projection.
// ---------------------------------------------------------------------------
__global__ __launch_bounds__(128) void gat_gemm64_wmma(
    const float* __restrict__ in,
    const float* __restrict__ W,
    float* __restrict__ out)
{
    __shared__ float xs[16 * XS_STRIDE];

    const int row0 = blockIdx.x * 16;
    const int t    = threadIdx.x;

    {
        const int r = t >> 3;
        const int c = (t & 7) * 8;
        const float4* src = reinterpret_cast<const float4*>(in + (row0 + r) * GAT_C + c);
        float4 a0 = src[0];
        float4 a1 = src[1];
        *reinterpret_cast<float4*>(&xs[r * XS_STRIDE + c])     = a0;
        *reinterpret_cast<float4*>(&xs[r * XS_STRIDE + c + 4]) = a1;
    }
    __syncthreads();

    const int wave = t >> 5;
    const int lane = t & 31;
    const int n0   = wave * 16;

    const int arow = lane & 15;
    const int koff = (lane >> 4) * 2;
    const int bn   = n0 + (lane & 15);
    const int bk   = (lane >> 4) * 2;

    v8f acc = {};
#pragma unroll
    for (int kk = 0; kk < 16; ++kk) {
        v2f a, b;
        a.x = xs[arow * XS_STRIDE + kk * 4 + koff];
        a.y = xs[arow * XS_STRIDE + kk * 4 + koff + 1];
        b.x = W[(kk * 4 + bk) * GAT_C + bn];
        b.y = W[(kk * 4 + bk + 1) * GAT_C + bn];
        acc = __builtin_amdgcn_wmma_f32_16x16x4_f32(
            false, a, false, b, (short)0, acc, false, false);
    }

    const int om = (lane >> 4) * 8;
    const int on = lane & 15;
#pragma unroll
    for (int r = 0; r < 8; ++r)
        out[(row0 + om + r) * GAT_C + n0 + on] = acc[r];
}

// ---------------------------------------------------------------------------
// Edge phase: 16 lanes per edge, lane `sub` owns the 4 contiguous channels
// [4*sub, 4*sub+4), all belonging to head sub>>2.
//   - 3x global_load_b128 per lane (q/k/v rows), L2-resident gathers
//   - head dot product: dot4 in-lane + 2 shuffles (xor 1,2 within the head's
//     4-lane subgroup)
//   - softmax across the 4 heads: 4 shuffles (xor 4,8 across head subgroups)
//   - scatter: 4 global_atomic_add_f32 per lane
// Each group handles EPG consecutive edges, software-pipelined with
// global_prefetch of the next edge's q/k/v rows to hide L2 latency.
// ---------------------------------------------------------------------------
#define EPG 4  // edges per 16-lane group; GAT_E % EPG == 0

__global__ __launch_bounds__(256) void gat_edge_kernel(
    const int* __restrict__ edge_index,   // [2, E]
    const float* __restrict__ q,
    const float* __restrict__ k,
    const float* __restrict__ v,
    float* __restrict__ out_n)
{
    const long long gtid = (long long)blockIdx.x * blockDim.x + threadIdx.x;
    const int group = (int)(gtid >> 4);
    const int sub   = (int)(gtid & 15);
    const int c0    = sub * 4;            // channel base; head = sub >> 2

    const int e0 = group * EPG;
    int row = edge_index[e0];             // destination (q side, scatter target)
    int col = edge_index[GAT_E + e0];     // source (k/v side)

#pragma unroll
    for (int i = 0; i < EPG; ++i) {
        int nrow = 0, ncol = 0;
        if (i + 1 < EPG) {                // uniform across the wave
            nrow = edge_index[e0 + i + 1];
            ncol = edge_index[GAT_E + e0 + i + 1];
            __builtin_prefetch(q + (long long)nrow * GAT_C + c0, 0, 3);
            __builtin_prefetch(k + (long long)ncol * GAT_C + c0, 0, 3);
            __builtin_prefetch(v + (long long)ncol * GAT_C + c0, 0, 3);
        }

        const float4 q4 = *reinterpret_cast<const float4*>(q + (long long)row * GAT_C + c0);
        const float4 k4 = *reinterpret_cast<const float4*>(k + (long long)col * GAT_C + c0);
        const float4 v4 = *reinterpret_cast<const float4*>(v + (long long)col * GAT_C + c0);

        // per-head dot product (D=16 spread over 4 lanes x 4 channels)
        float p = q4.x * k4.x + q4.y * k4.y + q4.z * k4.z + q4.w * k4.w;
        p += __shfl_xor(p, 1, 16);
        p += __shfl_xor(p, 2, 16);
        p *= 0.25f;                               // scale = D^-0.5
        const float s = (p > 0.0f) ? p : 0.2f * p; // leaky_relu

        // softmax over the 4 heads (heads live in 4-lane subgroups)
        float m = fmaxf(s, __shfl_xor(s, 4, 16));
        m = fmaxf(m, __shfl_xor(m, 8, 16));
        const float eh = __expf(s - m);
        float t = eh + __shfl_xor(eh, 4, 16);
        t += __shfl_xor(t, 8, 16);
        const float w = eh / t;

        float* __restrict__ dst = out_n + (long long)row * GAT_C + c0;
        atomicAdd(dst + 0, w * v4.x);
        atomicAdd(dst + 1, w * v4.y);
        atomicAdd(dst + 2, w * v4.z);
        atomicAdd(dst + 3, w * v4.w);

        row = nrow;
        col = ncol;
    }
}

// ---------------------------------------------------------------------------
__global__ __launch_bounds__(256) void gat_zero_kernel(float4* __restrict__ p, int n4)
{
    int i = blockIdx.x * blockDim.x + threadIdx.x;
    if (i < n4) p[i] = make_float4(0.f, 0.f, 0.f, 0.f);
}

// ---------------------------------------------------------------------------
extern "C" void kernel_launch(void* const* d_in, const int* in_sizes, int n_in,
                              void* d_out, int out_size, void* d_ws, size_t ws_size,
                              hipStream_t stream) {
    const float* x   = (const float*)d_in[0];
    const int*   ei  = (const int*)d_in[1];
    const float* Wq  = (const float*)d_in[2];
    const float* Wk  = (const float*)d_in[3];
    const float* Wv  = (const float*)d_in[4];
    const float* Wo  = (const float*)d_in[5];
    float*       out = (float*)d_out;

    // workspace: q | k | v | out_n, each N*C floats (25.6 MB; 102.4 MB total)
    const long long NC = (long long)GAT_N * GAT_C;
    float* q     = (float*)d_ws;
    float* k     = q + NC;
    float* v     = k + NC;
    float* out_n = v + NC;

    // 1) zero the scatter accumulator
    {
        int n4 = (int)(NC / 4);
        gat_zero_kernel<<<(n4 + 255) / 256, 256, 0, stream>>>((float4*)out_n, n4);
    }
    // 2) fused QKV projection (WMMA f32)
    gat_qkv_wmma<<<GAT_N / 16, 128, 0, stream>>>(x, Wq, Wk, Wv, q, k, v);
    // 3) edge attention + scatter (16 lanes/edge, EPG edges per group)
    {
        long long threads = ((long long)GAT_E / EPG) * 16;
        gat_edge_kernel<<<(unsigned)(threads / 256), 256, 0, stream>>>(
            ei, q, k, v, out_n);
    }
    // 4) output projection (WMMA f32)
    gat_gemm64_wmma<<<GAT_N / 16, 128, 0, stream>>>(out_n, Wo, out);
}